// BLSTMScratch_69655779606698
// MI455X (gfx1250) — compile-verified
//
#include <hip/hip_runtime.h>
#include <hip/hip_bf16.h>

// ---------------------------------------------------------------------------
// BLSTM on MI455X (gfx1250). fp32 WMMA (V_WMMA_F32_16X16X4_F32) GEMMs for
// gates + projection; weights stay L2-resident (32MB << 192MB L2).
// Gates GEMM: 32x32 per-wave tile (4 accumulators); W panel staged K-major
// (transposed) in LDS so each B fragment is one contiguous ds_load_b64
// (no register-assembly movs), pad stride 68 -> conflict-free reads.
// ---------------------------------------------------------------------------

typedef __attribute__((ext_vector_type(2))) float v2f;
typedef __attribute__((ext_vector_type(8))) float v8f;

#define BB   128   // batch
#define SS   512   // sequence
#define HH   1024  // hidden
#define VV   128   // vocab
#define N4   4096  // 4*H

// ---------------------------------------------------------------------------
// gates(B,4H) = concat(emb[x[:,t]], h_in)(B,2H) @ W(2H,4H) + bias
// grid: 64 blocks; block = 8 waves covering a 128(M) x 64(N) tile.
// wave tile: 32(M) x 32(N) = 4 accumulators.
// ---------------------------------------------------------------------------
__global__ __launch_bounds__(256) void blstm_gates_gemm(
    const int*   __restrict__ x,     // (B,S)
    const float* __restrict__ emb,   // (V,H)
    const float* __restrict__ h_in,  // (B,H)
    const float* __restrict__ W,     // (2H,4H) row-major
    const float* __restrict__ bias,  // (4H)
    float*       __restrict__ gates, // (B,4H)
    int t)
{
    // K-major (transposed) W panel: [n-col 0..63][K 0..63], stride 68 pad.
    __shared__ float Bst[64][68];

    const int tid  = threadIdx.x;
    const int wave = tid >> 5;
    const int lane = tid & 31;
    const int l    = lane & 15;      // lane within half
    const int half = lane >> 4;      // A frag: K+=2 ; C frag: M+=8
    const int m0   = (wave & 3) * 32;      // M strip base (32 rows)
    const int nloc = (wave >> 2) * 32;     // local N base (0 or 32)
    const int nb0  = blockIdx.x * 64;      // block N base
    const int r0   = m0 + l;               // A row, frag 0
    const int r1   = m0 + 16 + l;          // A row, frag 1

    // fused embedding gather: per-lane A row pointers for the x-phase
    const int tok0 = x[r0 * SS + t];
    const int tok1 = x[r1 * SS + t];
    const float* e0 = emb  + (size_t)tok0 * HH;
    const float* e1 = emb  + (size_t)tok1 * HH;
    const float* h0 = h_in + (size_t)r0 * HH;
    const float* h1 = h_in + (size_t)r1 * HH;

    v8f acc00 = {}, acc01 = {}, acc10 = {}, acc11 = {};

    for (int kc = 0; kc < 2 * HH; kc += 64) {
        __syncthreads();
        // stage W[kc..kc+63][nb0..nb0+63] into LDS, transposed to K-major
        for (int i = tid; i < 64 * 16; i += 256) {
            const int kk = i >> 4;
            const int n4 = (i & 15) * 4;
            const float4 w = *(const float4*)(W + (size_t)(kc + kk) * N4 + nb0 + n4);
            Bst[n4 + 0][kk] = w.x;
            Bst[n4 + 1][kk] = w.y;
            Bst[n4 + 2][kk] = w.z;
            Bst[n4 + 3][kk] = w.w;
        }
        __syncthreads();

        const float* a0p = (kc < HH) ? (e0 + kc) : (h0 + (kc - HH));
        const float* a1p = (kc < HH) ? (e1 + kc) : (h1 + (kc - HH));

#pragma unroll
        for (int kk = 0; kk < 64; kk += 4) {
            // A 16x4 f32 frags: lanes 0-15 -> K=kk,kk+1 ; lanes 16-31 -> K=kk+2,kk+3
            v2f a0 = *(const v2f*)(a0p + kk + 2 * half);
            v2f a1 = *(const v2f*)(a1p + kk + 2 * half);
            // B 4x16 frags: contiguous b64 from K-major LDS tile
            v2f b0 = *(const v2f*)&Bst[nloc + l][kk + 2 * half];
            v2f b1 = *(const v2f*)&Bst[nloc + 16 + l][kk + 2 * half];
            acc00 = __builtin_amdgcn_wmma_f32_16x16x4_f32(false, a0, false, b0, (short)0, acc00, false, false);
            acc01 = __builtin_amdgcn_wmma_f32_16x16x4_f32(false, a0, false, b1, (short)0, acc01, false, false);
            acc10 = __builtin_amdgcn_wmma_f32_16x16x4_f32(false, a1, false, b0, (short)0, acc10, false, false);
            acc11 = __builtin_amdgcn_wmma_f32_16x16x4_f32(false, a1, false, b1, (short)0, acc11, false, false);
        }
    }

    const int   c0  = nb0 + nloc + l;
    const int   c1  = c0 + 16;
    const float bv0 = bias[c0];
    const float bv1 = bias[c1];
#pragma unroll
    for (int r = 0; r < 8; ++r) {
        const int mA = m0 + r + 8 * half;        // C layout: VGPR r holds M=r (+8 hi half)
        const int mB = m0 + 16 + r + 8 * half;
        gates[(size_t)mA * N4 + c0] = acc00[r] + bv0;
        gates[(size_t)mA * N4 + c1] = acc01[r] + bv1;
        gates[(size_t)mB * N4 + c0] = acc10[r] + bv0;
        gates[(size_t)mB * N4 + c1] = acc11[r] + bv1;
    }
}

// ---------------------------------------------------------------------------
// elementwise LSTM cell: c' = sig(f)*c + sig(i)*tanh(g); h' = sig(o)*tanh(c')
// optionally mirrors h' into h_store (backward-pass history buffer).
// ---------------------------------------------------------------------------
__device__ __forceinline__ float sigf(float v) { return 1.0f / (1.0f + expf(-v)); }

__global__ __launch_bounds__(256) void blstm_cell(
    const float* __restrict__ gates,   // (B,4H)
    float* __restrict__ h,             // (B,H) in/out
    float* __restrict__ c,             // (B,H) in/out
    float* __restrict__ h_store)       // (B,H) or nullptr
{
    const int i = blockIdx.x * blockDim.x + threadIdx.x;  // 0..B*H-1
    const int b = i >> 10;
    const int j = i & (HH - 1);
    const float* g = gates + (size_t)b * N4;
    const float ig = g[j];
    const float fg = g[HH + j];
    const float og = g[2 * HH + j];
    const float gg = g[3 * HH + j];
    const float cn = sigf(fg) * c[i] + sigf(ig) * tanhf(gg);
    const float hn = sigf(og) * tanhf(cn);
    c[i] = cn;
    h[i] = hn;
    if (h_store) h_store[i] = hn;
}

// ---------------------------------------------------------------------------
// logits[:,t,:] = concat(h_f, h_b_t)(B,2H) @ W_out(2H,V) + b_out
// grid: 8 blocks (N tiles), 8 waves/block (M tiles). K-major LDS staging.
// ---------------------------------------------------------------------------
__global__ __launch_bounds__(256) void blstm_proj_gemm(
    const float* __restrict__ hf,     // (B,H)
    const float* __restrict__ hb_t,   // (B,H)
    const float* __restrict__ Wout,   // (2H,V)
    const float* __restrict__ bout,   // (V)
    float*       __restrict__ logits, // (B,S,V)
    int t)
{
    __shared__ float Bst[16][68];    // [n-col 0..15][K 0..63]

    const int wave  = threadIdx.x >> 5;
    const int lane  = threadIdx.x & 31;
    const int l     = lane & 15;
    const int half  = lane >> 4;
    const int tileM = wave;          // 0..7
    const int tileN = blockIdx.x;    // 0..7
    const int m     = tileM * 16 + l;
    const int n0    = tileN * 16;

    const float* frow = hf   + (size_t)m * HH;
    const float* brow = hb_t + (size_t)m * HH;

    v8f acc = {};

    for (int kc = 0; kc < 2 * HH; kc += 64) {
        __syncthreads();
        // stage Wout[kc..kc+63][n0..n0+15] into LDS, transposed to K-major
        for (int i = threadIdx.x; i < 64 * 4; i += 256) {
            const int kk = i >> 2;
            const int n4 = (i & 3) * 4;
            const float4 w = *(const float4*)(Wout + (size_t)(kc + kk) * VV + n0 + n4);
            Bst[n4 + 0][kk] = w.x;
            Bst[n4 + 1][kk] = w.y;
            Bst[n4 + 2][kk] = w.z;
            Bst[n4 + 3][kk] = w.w;
        }
        __syncthreads();

        const float* aptr = (kc < HH) ? (frow + kc) : (brow + (kc - HH));

#pragma unroll
        for (int kk = 0; kk < 64; kk += 4) {
            v2f a = *(const v2f*)(aptr + kk + 2 * half);
            v2f b = *(const v2f*)&Bst[l][kk + 2 * half];
            acc = __builtin_amdgcn_wmma_f32_16x16x4_f32(
                false, a, false, b, (short)0, acc, false, false);
        }
    }

    const float bv = bout[n0 + l];
#pragma unroll
    for (int r = 0; r < 8; ++r) {
        const int mrow = tileM * 16 + r + 8 * half;
        logits[(size_t)mrow * SS * VV + (size_t)t * VV + n0 + l] = acc[r] + bv;
    }
}

// ---------------------------------------------------------------------------
// host-side orchestration (graph-capture safe: only kernel launches + async
// D2D copies on `stream`).
// ---------------------------------------------------------------------------
extern "C" void kernel_launch(void* const* d_in, const int* in_sizes, int n_in,
                              void* d_out, int out_size, void* d_ws, size_t ws_size,
                              hipStream_t stream) {
    (void)in_sizes; (void)n_in; (void)out_size; (void)ws_size;

    const int*   x     = (const int*)  d_in[0];
    const float* h_f0  = (const float*)d_in[1];
    const float* h_b0  = (const float*)d_in[2];
    const float* c_f0  = (const float*)d_in[3];
    const float* c_b0  = (const float*)d_in[4];
    const float* emb   = (const float*)d_in[5];
    const float* W_f   = (const float*)d_in[6];
    const float* b_f   = (const float*)d_in[7];
    const float* W_b   = (const float*)d_in[8];
    const float* b_b   = (const float*)d_in[9];
    const float* W_out = (const float*)d_in[10];
    const float* b_out = (const float*)d_in[11];

    float* out = (float*)d_out;
    float* ws  = (float*)d_ws;

    const size_t BH  = (size_t)BB * HH;            // 131072
    const size_t NBH = BH * sizeof(float);

    // workspace layout (floats)
    float* hS    = ws;                 // (B,H) running hidden state
    float* cS    = ws + BH;            // (B,H) running cell state
    float* gates = ws + 2 * BH;        // (B,4H)
    float* hball = ws + 2 * BH + (size_t)BB * N4;   // (S,B,H) backward outputs, 256MB

    // d_out layout: logits(B,S,V) | hf | hb | cf | cb
    float* logits = out;
    float* out_hf = out + (size_t)BB * SS * VV;
    float* out_hb = out_hf + BH;
    float* out_cf = out_hb + BH;
    float* out_cb = out_cf + BH;

    // ---------------- backward direction ----------------
    hipMemcpyAsync(hS, h_b0, NBH, hipMemcpyDeviceToDevice, stream);
    hipMemcpyAsync(cS, c_b0, NBH, hipMemcpyDeviceToDevice, stream);
    for (int t = SS - 1; t >= 0; --t) {
        blstm_gates_gemm<<<64, 256, 0, stream>>>(x, emb, hS, W_b, b_b, gates, t);
        blstm_cell<<<(int)(BH / 256), 256, 0, stream>>>(gates, hS, cS, hball + (size_t)t * BH);
    }
    hipMemcpyAsync(out_hb, hS, NBH, hipMemcpyDeviceToDevice, stream);
    hipMemcpyAsync(out_cb, cS, NBH, hipMemcpyDeviceToDevice, stream);

    // ---------------- forward direction + fused projection ----------------
    hipMemcpyAsync(hS, h_f0, NBH, hipMemcpyDeviceToDevice, stream);
    hipMemcpyAsync(cS, c_f0, NBH, hipMemcpyDeviceToDevice, stream);
    for (int t = 0; t < SS; ++t) {
        blstm_gates_gemm<<<64, 256, 0, stream>>>(x, emb, hS, W_f, b_f, gates, t);
        blstm_cell<<<(int)(BH / 256), 256, 0, stream>>>(gates, hS, cS, nullptr);
        blstm_proj_gemm<<<8, 256, 0, stream>>>(hS, hball + (size_t)t * BH, W_out, b_out, logits, t);
    }
    hipMemcpyAsync(out_hf, hS, NBH, hipMemcpyDeviceToDevice, stream);
    hipMemcpyAsync(out_cf, cS, NBH, hipMemcpyDeviceToDevice, stream);
}